// Net_89481348645168
// MI455X (gfx1250) — compile-verified
//
#include <hip/hip_runtime.h>
#include <hip/hip_bf16.h>
#include <math.h>

typedef __attribute__((ext_vector_type(2))) float v2f;
typedef __attribute__((ext_vector_type(8))) float v8f;
typedef __attribute__((ext_vector_type(2))) int   v2i;

#define N_STEPS 30

// ---------------------------------------------------------------------------
// Pass 1: histogram of destination rows
// ---------------------------------------------------------------------------
__global__ void count_rows_kernel(const long long* __restrict__ edge_index,
                                  int* __restrict__ counts, int E) {
    int e = blockIdx.x * blockDim.x + threadIdx.x;
    if (e < E) {
        int r = (int)edge_index[e];
        atomicAdd(&counts[r], 1);
    }
}

// ---------------------------------------------------------------------------
// Pass 2: single-block exclusive scan -> CSR offsets (and a scatter cursor)
// ---------------------------------------------------------------------------
__global__ __launch_bounds__(1024)
void scan_kernel(const int* __restrict__ counts,
                 int* __restrict__ offsets,
                 int* __restrict__ cursor, int n) {
    __shared__ int sh[1024];
    int carry = 0;  // uniform across the block
    for (int base = 0; base < n; base += 1024) {
        int i = base + (int)threadIdx.x;
        int v = (i < n) ? counts[i] : 0;
        sh[threadIdx.x] = v;
        __syncthreads();
        for (int off = 1; off < 1024; off <<= 1) {
            int t = (threadIdx.x >= (unsigned)off) ? sh[threadIdx.x - off] : 0;
            __syncthreads();
            sh[threadIdx.x] += t;
            __syncthreads();
        }
        int incl = sh[threadIdx.x];
        int btot = sh[1023];
        if (i < n) {
            int excl = carry + incl - v;
            offsets[i] = excl;
            cursor[i]  = excl;
        }
        __syncthreads();
        carry += btot;
    }
    if (threadIdx.x == 0) offsets[n] = carry;
}

// ---------------------------------------------------------------------------
// Pass 3: scatter edges into CSR order as interleaved 8-byte records:
//   .x = col (20 bits) | local_row (4 bits) << 20,  .y = weight bits
// ---------------------------------------------------------------------------
__global__ void scatter_edges_kernel(const long long* __restrict__ edge_index,
                                     const float* __restrict__ edge_attr,
                                     int* __restrict__ cursor,
                                     v2i* __restrict__ emeta, int E) {
    int e = blockIdx.x * blockDim.x + threadIdx.x;
    if (e >= E) return;
    int r = (int)edge_index[e];
    int c = (int)edge_index[(size_t)E + e];
    int p = atomicAdd(&cursor[r], 1);
    v2i rec;
    rec.x = c | ((r & 15) << 20);
    rec.y = __float_as_int(edge_attr[e]);
    emeta[p] = rec;
}

// ---------------------------------------------------------------------------
// Hot loop: one wave per 16-row tile. Segment-sum via V_WMMA_F32_16X16X4_F32:
//   A(16x4) = one-hot(dest row) * weight   (4 edges per WMMA)
//   B(4x16) = gathered x[col] rows         (coalesced 64B per half-wave)
//   C(16x16) accumulates 16 rows x 16 classes, no atomics.
// Scalar trip count (readfirstlane) -> uniform loop, EXEC all-1s at WMMA.
// 8 edges / 2 independent WMMA accumulators per iteration; edge meta as b64.
// ---------------------------------------------------------------------------
__global__ __launch_bounds__(128)
void spmm_step_kernel(const int* __restrict__ offsets,
                      const v2i* __restrict__ emeta,
                      const float* __restrict__ xin,
                      float* __restrict__ xout,
                      int ntiles, int nnodes) {
    int wave = blockIdx.x * (blockDim.x >> 5) + ((int)threadIdx.x >> 5);
    wave = __builtin_amdgcn_readfirstlane(wave);   // wave-uniform -> SGPR
    if (wave >= ntiles) return;                    // uniform branch
    int lane  = (int)threadIdx.x & 31;
    int m     = lane & 15;                         // row (A) / class (B,D)
    int khalf = lane >> 4;                         // which K-half this lane holds

    int rowlo = wave * 16;
    int s    = __builtin_amdgcn_readfirstlane(offsets[rowlo]);
    int eEnd = __builtin_amdgcn_readfirstlane(offsets[min(rowlo + 16, nnodes)]);

    v8f acc0 = {};
    v8f acc1 = {};
    const int kbase = 2 * khalf;
    const int last  = eEnd - 1;

    for (int j = s; j < eEnd; j += 8) {            // scalar loop bounds
        __builtin_prefetch(emeta + j + 256, 0, 1); // 2KB ahead in edge stream

        // group 0: edges j..j+3 ; group 1: edges j+4..j+7
        int e0 = j + kbase, e1 = e0 + 1, e2 = e0 + 4, e3 = e0 + 5;
        int i0 = min(e0, last), i1 = min(e1, last);
        int i2 = min(e2, last), i3 = min(e3, last);

        // unconditional clamped b64 loads (always in-bounds), zero via cndmask
        v2i r0 = emeta[i0], r1 = emeta[i1], r2 = emeta[i2], r3 = emeta[i3];
        float w0 = (e0 <= last) ? __int_as_float(r0.y) : 0.0f;
        float w1 = (e1 <= last) ? __int_as_float(r1.y) : 0.0f;
        float w2 = (e2 <= last) ? __int_as_float(r2.y) : 0.0f;
        float w3 = (e3 <= last) ? __int_as_float(r3.y) : 0.0f;

        v2f a0, b0, a1, b1;
        a0.x = (m == (r0.x >> 20)) ? w0 : 0.0f;    // A[m][2*khalf]
        a0.y = (m == (r1.x >> 20)) ? w1 : 0.0f;    // A[m][2*khalf+1]
        a1.x = (m == (r2.x >> 20)) ? w2 : 0.0f;
        a1.y = (m == (r3.x >> 20)) ? w3 : 0.0f;
        b0.x = xin[(r0.x & 0xFFFFF) * 16 + m];     // B[2*khalf][m]
        b0.y = xin[(r1.x & 0xFFFFF) * 16 + m];     // B[2*khalf+1][m]
        b1.x = xin[(r2.x & 0xFFFFF) * 16 + m];
        b1.y = xin[(r3.x & 0xFFFFF) * 16 + m];

        acc0 = __builtin_amdgcn_wmma_f32_16x16x4_f32(
            false, a0, false, b0, (short)0, acc0, false, false);
        acc1 = __builtin_amdgcn_wmma_f32_16x16x4_f32(
            false, a1, false, b1, (short)0, acc1, false, false);
    }

    acc0 = acc0 + acc1;

    // D layout: VGPR v -> row = rowlo + v + 8*khalf, class = m
    int rowbase = rowlo + 8 * khalf;
    if (rowlo + 16 <= nnodes) {                    // uniform fast path
#pragma unroll
        for (int v = 0; v < 8; ++v)
            xout[(rowbase + v) * 16 + m] = acc0[v];
    } else {
#pragma unroll
        for (int v = 0; v < 8; ++v)
            if (rowbase + v < nnodes)
                xout[(rowbase + v) * 16 + m] = acc0[v];
    }
}

// ---------------------------------------------------------------------------
// Final: log-softmax over 16 classes; 16-lane shuffle reductions (wave32)
// ---------------------------------------------------------------------------
__global__ void log_softmax_kernel(const float* __restrict__ x,
                                   float* __restrict__ out, int n16) {
    int t = blockIdx.x * blockDim.x + threadIdx.x;
    if (t >= n16) return;
    float v = x[t];
    float mx = v;
    for (int off = 1; off < 16; off <<= 1)
        mx = fmaxf(mx, __shfl_xor(mx, off, 16));
    float s = __expf(v - mx);
    for (int off = 1; off < 16; off <<= 1)
        s += __shfl_xor(s, off, 16);
    out[t] = (v - mx) - __logf(s);
}

// ---------------------------------------------------------------------------
extern "C" void kernel_launch(void* const* d_in, const int* in_sizes, int n_in,
                              void* d_out, int out_size, void* d_ws, size_t ws_size,
                              hipStream_t stream) {
    const long long* edge_index = (const long long*)d_in[0]; // (2, E) int64
    const float*     edge_attr  = (const float*)d_in[1];     // (E,)  f32
    const float*     one_hot    = (const float*)d_in[2];     // (N,16) f32
    float* out = (float*)d_out;

    int E = in_sizes[1];
    int N = in_sizes[2] / 16;

    // workspace carve-out (256B aligned slices)
    char* ws = (char*)d_ws;
    auto carve = [&](size_t bytes) {
        char* p = ws;
        ws += (bytes + 255) & ~(size_t)255;
        return p;
    };
    int* offsets = (int*)carve((size_t)(N + 1) * 4);
    int* cursor  = (int*)carve((size_t)N * 4);
    int* counts  = (int*)carve((size_t)N * 4);
    v2i* emeta   = (v2i*)carve((size_t)E * 8);
    float* x0    = (float*)carve((size_t)N * 16 * 4);
    float* x1    = (float*)carve((size_t)N * 16 * 4);

    // CSR build (amortized over 30 SpMM steps)
    hipMemsetAsync(counts, 0, (size_t)N * 4, stream);
    count_rows_kernel<<<(E + 255) / 256, 256, 0, stream>>>(edge_index, counts, E);
    scan_kernel<<<1, 1024, 0, stream>>>(counts, offsets, cursor, N);
    scatter_edges_kernel<<<(E + 255) / 256, 256, 0, stream>>>(
        edge_index, edge_attr, cursor, emeta, E);

    // 30 propagation steps, ping-pong buffers (step 0 reads one_hot directly)
    int ntiles = (N + 15) / 16;
    int nblk   = (ntiles + 3) / 4;   // 4 waves (128 threads) per block
    const float* xin = one_hot;
    float* bufs[2] = {x0, x1};
    for (int st = 0; st < N_STEPS; ++st) {
        float* xo = bufs[st & 1];
        spmm_step_kernel<<<nblk, 128, 0, stream>>>(offsets, emeta,
                                                   xin, xo, ntiles, N);
        xin = xo;
    }

    log_softmax_kernel<<<(N * 16 + 255) / 256, 256, 0, stream>>>(xin, out, N * 16);
}